// TnoFDNoRPE_84679575208688
// MI455X (gfx1250) — compile-verified
//
#include <hip/hip_runtime.h>

// ---------------------------------------------------------------------------
// TNO frequency-domain causal depthwise conv (B=8, N=8192, D=512) for gfx1250.
// FFT convolution: full 16384-pt complex FFTs in LDS (320KB/WGP), radix-16
// Stockham stages on the matrix pipes via V_WMMA_F32_16X16X4_F32, channel
// pairs packed into one complex sequence. All LDS accesses via addrspace(3)
// offsets (no generic pointers -> no null checks / 64-bit address math).
// ---------------------------------------------------------------------------

#define FFT_N   16384
#define HALF_N  8192
#define NBINS   8193          // fft_size/2 + 1
#define DIM     512
#define SEQ     8192
#define BATCH   8
#define NFREQ   512
#define TPB     512           // 16 wave32 waves; 1 WG/WGP (LDS-limited)
#define WAVES   (TPB / 32)
#define BPW     (64 / WAVES)  // radix-16 butterfly batches per wave

typedef float v2f __attribute__((ext_vector_type(2)));
typedef float v8f __attribute__((ext_vector_type(8)));

struct __align__(8) cpx { float re, im; };

__device__ __forceinline__ cpx cmul(cpx a, cpx b) {
  cpx r; r.re = a.re * b.re - a.im * b.im; r.im = a.re * b.im + a.im * b.re; return r;
}

// ---------------------------------------------------------------------------
// One Stockham radix-16 stage over the 16384-pt sequence (offsets into LDS).
//   s = 1<<logs, n = FFT_N/s. Butterfly u = p*s + q: gather lds[in + q + s*p
//   + 1024*K], b = DFT16(a), scatter lds[out + q + s*(16p + M)] * w_n^{p*M}.
// 16 butterflies form the 16-column B matrix; DFT16 is A; complex product =
// 4 real 16x16x16 matmuls = 16 x WMMA f32 16x16x4. F32 WMMA has no A/B NEG
// modifier (NEG = {CNeg,0,0}), so -Wi uses a pre-negated fragment.
// EXEC all-ones: full block, uniform control flow.
// ---------------------------------------------------------------------------
template<int SIGN>
__device__ __forceinline__ void radix16_stage(int inOff, int outOff,
                                              int logs, int lane, int wave)
{
  extern __shared__ cpx lds[];
  const int   s    = 1 << logs;
  const int   n    = FFT_N >> logs;
  const int   col  = lane & 15;      // A: M row / B,C: N column
  const int   half = lane >> 4;
  const float W16  = (float)SIGN * 6.28318530717958647692f * (1.0f / 16.0f);

  // A fragments: W[M][K] = exp(SIGN*2*pi*i*M*K/16).
  // 16x4 f32 A layout: lane&15 = M, VGPR t holds K = 4q + t + 2*(lane>>4).
  v2f Ar[4], Ai[4], AiN[4];
#pragma unroll
  for (int qq = 0; qq < 4; ++qq) {
#pragma unroll
    for (int t = 0; t < 2; ++t) {
      const int K = (qq << 2) + t + (half << 1);
      float sA, cA;
      __sincosf(W16 * (float)((col * K) & 15), &sA, &cA);
      Ar[qq][t] = cA; Ai[qq][t] = sA; AiN[qq][t] = -sA;
    }
  }

  // 64 butterfly batches per stage, BPW per wave.
  for (int it = 0; it < BPW; ++it) {
    const int g = wave * BPW + it;         // batch 0..63
    const int u = (g << 4) + col;          // this lane's butterfly/column
    const int q = u & (s - 1);
    const int p = u >> logs;
    const int base_in = inOff + q + (p << logs);  // K-stride = FFT_N/16 = 1024

    v2f Br[4], Bi[4];
#pragma unroll
    for (int qq = 0; qq < 4; ++qq) {
#pragma unroll
      for (int t = 0; t < 2; ++t) {
        const int K = (qq << 2) + t + (half << 1);
        const cpx v = lds[base_in + (K << 10)];
        Br[qq][t] = v.re; Bi[qq][t] = v.im;
      }
    }

    v8f Cr = {}, Ci = {};
#pragma unroll
    for (int qq = 0; qq < 4; ++qq) {
      // Cr += Wr*Br + (-Wi)*Bi ; Ci += Wr*Bi + Wi*Br
      Cr = __builtin_amdgcn_wmma_f32_16x16x4_f32(false, Ar[qq],  false, Br[qq],
                                                 (short)0, Cr, false, false);
      Cr = __builtin_amdgcn_wmma_f32_16x16x4_f32(false, AiN[qq], false, Bi[qq],
                                                 (short)0, Cr, false, false);
      Ci = __builtin_amdgcn_wmma_f32_16x16x4_f32(false, Ar[qq],  false, Bi[qq],
                                                 (short)0, Ci, false, false);
      Ci = __builtin_amdgcn_wmma_f32_16x16x4_f32(false, Ai[qq],  false, Br[qq],
                                                 (short)0, Ci, false, false);
    }

    // Post-DFT twiddle w_n^{p*M} and scatter. C layout: lane&15 = column,
    // VGPR r holds M = r + 8*(lane>>4).
    const int   base_out = outOff + q + (p << (logs + 4));
    const float wstep    = (float)SIGN * 6.28318530717958647692f *
                           (float)p / (float)n;
#pragma unroll
    for (int r = 0; r < 8; ++r) {
      const int M = r + (half << 3);
      float sw, cw;
      __sincosf(wstep * (float)M, &sw, &cw);
      cpx y;
      y.re = Cr[r] * cw - Ci[r] * sw;
      y.im = Cr[r] * sw + Ci[r] * cw;
      lds[base_out + (M << logs)] = y;
    }
  }
}

// Final radix-4 stage: s = 4096, n = 4, p = 0 (twiddle-free), pure VALU.
template<int SIGN>
__device__ __forceinline__ void radix4_stage(int inOff, int outOff,
                                             float scaleVal, int tid)
{
  extern __shared__ cpx lds[];
#pragma unroll
  for (int k = tid; k < 4096; k += TPB) {
    const int i = inOff + k, o = outOff + k;
    const cpx a0 = lds[i], a1 = lds[i + 4096], a2 = lds[i + 8192], a3 = lds[i + 12288];
    cpx t0, t1, t2, d, t3;
    t0.re = a0.re + a2.re; t0.im = a0.im + a2.im;
    t1.re = a0.re - a2.re; t1.im = a0.im - a2.im;
    t2.re = a1.re + a3.re; t2.im = a1.im + a3.im;
    d.re  = a1.re - a3.re; d.im  = a1.im - a3.im;
    t3.re = -(float)SIGN * d.im; t3.im = (float)SIGN * d.re;   // d * (SIGN*i)
    lds[o        ] = cpx{(t0.re + t2.re) * scaleVal, (t0.im + t2.im) * scaleVal};
    lds[o +  4096] = cpx{(t1.re + t3.re) * scaleVal, (t1.im + t3.im) * scaleVal};
    lds[o +  8192] = cpx{(t0.re - t2.re) * scaleVal, (t0.im - t2.im) * scaleVal};
    lds[o + 12288] = cpx{(t1.re - t3.re) * scaleVal, (t1.im - t3.im) * scaleVal};
  }
}

// Full 16384-pt complex FFT on LDS offsets. Input at off0, result at off0.
// SIGN = -1 forward, +1 inverse (pass 1/N scale for inverse).
template<int SIGN>
__device__ void fft16384(int off0, int off1, float scaleVal)
{
  const int tid  = threadIdx.x;
  const int lane = tid & 31;
  const int wave = tid >> 5;
  __syncthreads();
  radix16_stage<SIGN>(off0, off1, 0, lane, wave); __syncthreads();
  radix16_stage<SIGN>(off1, off0, 4, lane, wave); __syncthreads();
  radix16_stage<SIGN>(off0, off1, 8, lane, wave); __syncthreads();
  radix4_stage<SIGN>(off1, off0, scaleVal, tid);  __syncthreads();
}

// ---------------------------------------------------------------------------
// Kernel 1: causal spectrum S[d][f], f = 0..8192, per channel pair.
// a_d = [dc_d, interp(coeffs_d, 8192)]; t = irfft(a); k = t * w_analytic;
// S = rfft(k). Two real channels ride one complex FFT.
// ---------------------------------------------------------------------------
__global__ void __launch_bounds__(TPB, 1)
tno_spec_kernel(const float* __restrict__ coeffs, const float* __restrict__ dc,
                cpx* __restrict__ spec)
{
  extern __shared__ cpx lds[];
  const int tid = threadIdx.x;
  const int cp  = blockIdx.x;        // 0..255
  const int d0  = cp << 1;
  const int d1  = d0 + 1;

  // Packed real-even spectrum (half-pixel linear interp, in_len=512).
  for (int f = tid; f <= HALF_N; f += TPB) {
    float v0, v1;
    if (f == 0) {
      v0 = dc[d0]; v1 = dc[d1];
    } else {
      float src = ((float)(f - 1) + 0.5f) * ((float)NFREQ / (float)HALF_N) - 0.5f;
      src = fminf(fmaxf(src, 0.0f), (float)(NFREQ - 1));
      const int   lo = (int)floorf(src);
      const int   hi = min(lo + 1, NFREQ - 1);
      const float w  = src - (float)lo;
      v0 = coeffs[d0 * NFREQ + lo] * (1.0f - w) + coeffs[d0 * NFREQ + hi] * w;
      v1 = coeffs[d1 * NFREQ + lo] * (1.0f - w) + coeffs[d1 * NFREQ + hi] * w;
    }
    lds[f] = cpx{v0, v1};
    if (f > 0 && f < HALF_N) lds[FFT_N - f] = cpx{v0, v1};   // real-even mirror
  }

  // irfft -> even real time-kernel pair (re = ch d0, im = ch d1)
  fft16384<+1>(0, FFT_N, 1.0f / (float)FFT_N);

  // analytic-signal window: w[0]=1, w[1..8191]=2, w[8192]=1, else 0
  for (int m = tid; m < FFT_N; m += TPB) {
    const float w = (m == 0 || m == HALF_N) ? 1.0f : (m < HALF_N ? 2.0f : 0.0f);
    lds[m].re *= w; lds[m].im *= w;
  }

  // rfft of the (packed) causal kernels
  fft16384<-1>(0, FFT_N, 1.0f);

  // Unpack two real-signal spectra from packed transform G:
  // S0 = (G[f] + conj(G[-f]))/2 ; S1 = -i*(G[f] - conj(G[-f]))/2
  for (int f = tid; f <= HALF_N; f += TPB) {
    const cpx g1 = lds[f];
    const cpx g2 = lds[(FFT_N - f) & (FFT_N - 1)];
    cpx s0, s1;
    s0.re = 0.5f * (g1.re + g2.re);
    s0.im = 0.5f * (g1.im - g2.im);
    const float dr = g1.re - g2.re;
    const float di = g1.im + g2.im;
    s1.re = 0.5f * di;
    s1.im = -0.5f * dr;
    spec[(size_t)d0 * NBINS + f] = s0;
    spec[(size_t)d1 * NBINS + f] = s1;
  }
}

// ---------------------------------------------------------------------------
// Kernel 2: per (batch, channel-pair) FFT convolution.
//   z = x[:,d0] + i*x[:,d1] (zero-padded) -> Z = FFT(z)
//   unpack X0,X1, multiply by S0,S1, repack Y = Y0 + i*Y1 (conj-sym mirror)
//   y = IFFT(Y); out[:,d0] = Re(y)[:8192], out[:,d1] = Im(y)[:8192]
// ---------------------------------------------------------------------------
__global__ void __launch_bounds__(TPB, 1)
tno_fdconv_kernel(const float* __restrict__ x, const cpx* __restrict__ spec,
                  float* __restrict__ out)
{
  extern __shared__ cpx lds[];
  const int tid = threadIdx.x;
  const int b   = blockIdx.x >> 8;       // 256 channel pairs per batch
  const int cp  = blockIdx.x & 255;
  const int d0  = cp << 1;

  const cpx* S0 = spec + (size_t)d0 * NBINS;
  const cpx* S1 = S0 + NBINS;

  // Warm L2->WGP path for the spectrum while we stream x (consumed later).
  for (int f = tid * 16; f <= HALF_N; f += TPB * 16) {
    __builtin_prefetch(S0 + f, 0, 1);
    __builtin_prefetch(S1 + f, 0, 1);
  }

  // Load channel pair as complex; D-innermost layout makes (d,d+1) one b64.
  const float* xb = x + (size_t)b * SEQ * DIM + d0;
  for (int n = tid; n < SEQ; n += TPB)
    lds[n] = *(const cpx*)(xb + (size_t)n * DIM);
  for (int n = SEQ + tid; n < FFT_N; n += TPB)
    lds[n] = cpx{0.0f, 0.0f};

  fft16384<-1>(0, FFT_N, 1.0f);          // Z in lds[0..]

  // Spectral multiply into the second buffer (out-of-place + explicit mirror).
  for (int f = tid; f <= HALF_N; f += TPB) {
    const cpx z1 = lds[f];
    const cpx z2 = lds[(FFT_N - f) & (FFT_N - 1)];
    cpx X0, X1;
    X0.re = 0.5f * (z1.re + z2.re);
    X0.im = 0.5f * (z1.im - z2.im);
    const float dr = z1.re - z2.re;
    const float di = z1.im + z2.im;
    X1.re = 0.5f * di;
    X1.im = -0.5f * dr;
    const cpx Y0 = cmul(X0, S0[f]);
    const cpx Y1 = cmul(X1, S1[f]);
    lds[FFT_N + f] = cpx{Y0.re - Y1.im, Y0.im + Y1.re};
    if (f > 0 && f < HALF_N)
      lds[FFT_N + FFT_N - f] = cpx{Y0.re + Y1.im, -Y0.im + Y1.re};
  }

  fft16384<+1>(FFT_N, 0, 1.0f / (float)FFT_N);   // y in lds[FFT_N..]

  float* ob = out + (size_t)b * SEQ * DIM + d0;
  for (int n = tid; n < SEQ; n += TPB) {
    const cpx y = lds[FFT_N + n];
    *(cpx*)(ob + (size_t)n * DIM) = y;
  }
}

// ---------------------------------------------------------------------------
extern "C" void kernel_launch(void* const* d_in, const int* in_sizes, int n_in,
                              void* d_out, int out_size, void* d_ws, size_t ws_size,
                              hipStream_t stream)
{
  (void)in_sizes; (void)n_in; (void)out_size; (void)ws_size;
  const float* x      = (const float*)d_in[0];
  const float* coeffs = (const float*)d_in[1];
  const float* dc     = (const float*)d_in[2];
  float*       out    = (float*)d_out;
  cpx*         spec   = (cpx*)d_ws;     // needs 512 * 8193 * 8B = 33.6 MB

  const size_t ldsBytes = (size_t)2 * FFT_N * sizeof(cpx);   // 256 KB / WGP

  // Allow >64KB dynamic LDS (gfx1250 WGP has 320KB). Idempotent, no stream use.
  (void)hipFuncSetAttribute((const void*)tno_spec_kernel,
                            hipFuncAttributeMaxDynamicSharedMemorySize,
                            (int)ldsBytes);
  (void)hipFuncSetAttribute((const void*)tno_fdconv_kernel,
                            hipFuncAttributeMaxDynamicSharedMemorySize,
                            (int)ldsBytes);

  // 1) causal spectra for all 512 channels (256 channel pairs)
  tno_spec_kernel<<<DIM / 2, TPB, ldsBytes, stream>>>(coeffs, dc, spec);
  // 2) FFT convolution: 8 batches x 256 channel pairs
  tno_fdconv_kernel<<<BATCH * (DIM / 2), TPB, ldsBytes, stream>>>(x, spec, out);
}